// InteractionBlock_49168785605360
// MI455X (gfx1250) — compile-verified
//
#include <hip/hip_runtime.h>

typedef float v2f __attribute__((ext_vector_type(2)));
typedef float v4f __attribute__((ext_vector_type(4)));
typedef float v8f __attribute__((ext_vector_type(8)));

#define THREADS 256
#define ROWS_PB 128          // rows per block (8 waves x 16-row tiles)

// h = [x | d_chi] @ W + b ; a1 = h[:,:128] ; chi_out = h[:,128:][:,seg]*chi
__global__ __launch_bounds__(THREADS) void interaction_kernel(
    const float* __restrict__ x,      // (n,128)
    const float* __restrict__ chi,    // (n,16)
    const float* __restrict__ W,      // (132,132) row-major
    const float* __restrict__ b,      // (132,)
    float* __restrict__ a1,           // (n,128)
    float* __restrict__ chi_out,      // (n,16)
    int n)
{
    // LDS: Wt padded to 144 N-rows (zeros beyond 132), y tile (reused for h),
    // bias, b1 staging buffer
    __shared__ float wt[144 * 132];   // Wt[nn][k] = W[k][nn]
    __shared__ float ys[ROWS_PB * 132];
    __shared__ float bsh[132];
    __shared__ float hb[ROWS_PB * 4];

    const int  t    = threadIdx.x;
    const long row0 = (long)blockIdx.x * ROWS_PB;

    // ---- stage W transposed, vectorized global reads (33 float4 per row) ----
    for (int idx4 = t; idx4 < 33 * 132; idx4 += THREADS) {
        int k   = idx4 / 33;
        int nn0 = (idx4 - k * 33) * 4;
        v4f wv = *(const v4f*)(W + k * 132 + nn0);
        wt[(nn0 + 0) * 132 + k] = wv.x;
        wt[(nn0 + 1) * 132 + k] = wv.y;
        wt[(nn0 + 2) * 132 + k] = wv.z;
        wt[(nn0 + 3) * 132 + k] = wv.w;
    }
    for (int idx = t; idx < 12 * 132; idx += THREADS)
        wt[132 * 132 + idx] = 0.0f;          // zero-pad N = 132..143
    if (t < 132) bsh[t] = b[t];

    // ---- stage x tile as float4 (ys stride 132 keeps 16B alignment) ----
    for (int idx4 = t; idx4 < ROWS_PB * 32; idx4 += THREADS) {
        int r = idx4 >> 5, c4 = (idx4 & 31) << 2;
        long gr = row0 + r;
        v4f xv = (gr < n) ? *(const v4f*)(x + gr * 128 + c4) : (v4f)0.0f;
        *(v4f*)(ys + r * 132 + c4) = xv;
    }
    {   // d_chi: 2 threads per row, float4 chi reads
        int r = t >> 1, half = t & 1;
        long gr = row0 + r;
        const float* cr = chi + gr * 16;
        if (half == 0) {
            float s0 = 0.0f, s1 = 0.0f;
            if (gr < n) {
                v4f c = *(const v4f*)(cr);
                s0 = c.x * c.x;
                s1 = c.y * c.y + c.z * c.z + c.w * c.w;
            }
            ys[r * 132 + 128] = s0;
            ys[r * 132 + 129] = s1;
        } else {
            float s2 = 0.0f, s3 = 0.0f;
            if (gr < n) {
                v4f c1 = *(const v4f*)(cr + 4);
                v4f c2 = *(const v4f*)(cr + 8);
                v4f c3 = *(const v4f*)(cr + 12);
                s2 = c1.x * c1.x + c1.y * c1.y + c1.z * c1.z + c1.w * c1.w + c2.x * c2.x;
                s3 = c2.y * c2.y + c2.z * c2.z + c2.w * c2.w +
                     c3.x * c3.x + c3.y * c3.y + c3.z * c3.z + c3.w * c3.w;
            }
            ys[r * 132 + 130] = s2;
            ys[r * 132 + 131] = s3;
        }
    }
    __syncthreads();

    // ---- GEMM: wave w -> rows [16w,16w+16), 9 col tiles, K=132 in 33 steps ----
    const int w    = t >> 5;
    const int l    = t & 31;
    const int nloc = l & 15;                 // M for A, N for B, col within D
    const int koff = (l < 16) ? 0 : 2;       // 16x16x4 f32 A/B lane K-offset

    v8f acc[9];
    #pragma unroll
    for (int ct = 0; ct < 9; ++ct) acc[ct] = (v8f)0.0f;

    const float* arow = ys + (16 * w + nloc) * 132 + koff;
    const float* brow = wt + nloc * 132 + koff;

    for (int ks = 0; ks < 33; ++ks) {
        const int k0 = ks * 4;
        v2f afrag = *(const v2f*)(arow + k0);
        #pragma unroll
        for (int ct = 0; ct < 9; ++ct) {
            v2f bfrag = *(const v2f*)(brow + ct * (16 * 132) + k0);
            acc[ct] = __builtin_amdgcn_wmma_f32_16x16x4_f32(
                false, afrag, false, bfrag, (short)0, acc[ct], false, false);
        }
    }

    // ---- write h (+bias) back into ys (each wave owns its 16 rows; the wave
    //      has already consumed its A fragments, so no barrier needed here) ----
    const int rhalf = (l >> 4) << 3;         // lanes 16-31 hold rows M+8
    #pragma unroll
    for (int ct = 0; ct < 8; ++ct) {
        float bb = bsh[16 * ct + nloc];
        #pragma unroll
        for (int v = 0; v < 8; ++v) {
            int rloc = 16 * w + rhalf + v;
            ys[rloc * 132 + 16 * ct + nloc] = acc[ct][v] + bb;
        }
    }
    // ---- col tile 8: cols 128..131 -> b1 into LDS ----
    if (nloc < 4) {
        float bb = bsh[128 + nloc];
        #pragma unroll
        for (int v = 0; v < 8; ++v) {
            int rloc = 16 * w + rhalf + v;
            hb[rloc * 4 + nloc] = acc[8][v] + bb;
        }
    }
    __syncthreads();

    // ---- stream a1 out fully coalesced as float4 ----
    for (int idx4 = t; idx4 < ROWS_PB * 32; idx4 += THREADS) {
        int r = idx4 >> 5, c4 = (idx4 & 31) << 2;
        long gr = row0 + r;
        if (gr < n) {
            v4f hv = *(const v4f*)(ys + r * 132 + c4);
            *(v4f*)(a1 + gr * 128 + c4) = hv;
        }
    }

    // ---- chi_out = b1[:, seg] * chi, float4 ----
    for (int idx4 = t; idx4 < ROWS_PB * 4; idx4 += THREADS) {
        int r = idx4 >> 2, m0 = (idx4 & 3) << 2;
        long gr = row0 + r;
        if (gr < n) {
            v4f cv = *(const v4f*)(chi + gr * 16 + m0);
            v4f ov;
            {
                int m = m0 + 0, s = (m > 0) + (m > 3) + (m > 8);
                ov.x = hb[r * 4 + s] * cv.x;
            }
            {
                int m = m0 + 1, s = (m > 0) + (m > 3) + (m > 8);
                ov.y = hb[r * 4 + s] * cv.y;
            }
            {
                int m = m0 + 2, s = (m > 0) + (m > 3) + (m > 8);
                ov.z = hb[r * 4 + s] * cv.z;
            }
            {
                int m = m0 + 3, s = (m > 0) + (m > 3) + (m > 8);
                ov.w = hb[r * 4 + s] * cv.w;
            }
            *(v4f*)(chi_out + gr * 16 + m0) = ov;
        }
    }
}

extern "C" void kernel_launch(void* const* d_in, const int* in_sizes, int n_in,
                              void* d_out, int out_size, void* d_ws, size_t ws_size,
                              hipStream_t stream) {
    const float* x   = (const float*)d_in[0];
    const float* chi = (const float*)d_in[1];
    // d_in[2] = point_mask (unused by the reference math)
    const float* W   = (const float*)d_in[3];
    const float* b   = (const float*)d_in[4];
    const int n = in_sizes[2];                       // point_mask count == n

    float* a1      = (float*)d_out;
    float* chi_out = a1 + (size_t)n * 128;

    const int blocks = (n + ROWS_PB - 1) / ROWS_PB;
    hipLaunchKernelGGL(interaction_kernel, dim3(blocks), dim3(THREADS), 0, stream,
                       x, chi, W, b, a1, chi_out, n);
}